// SmoothnessLoss_59425167508112
// MI455X (gfx1250) — compile-verified
//
#include <hip/hip_runtime.h>

// CDNA5 / gfx1250, wave32.
typedef __attribute__((ext_vector_type(2))) float v2f;
typedef __attribute__((ext_vector_type(4))) float fv4;
typedef __attribute__((ext_vector_type(8))) float v8f;

#define H_DIM   4096
#define W_DIM   4096
#define HW_MASK (H_DIM * W_DIM - 1)     // per-channel flat mask (power of two)
#define NVEC4   ((2 * H_DIM * W_DIM) / 4)
#define INV_HW  (1.0f / (float)(H_DIM * W_DIM))

// ---------------------------------------------------------------------------
// Pass 1: streaming pass. Contribution per element: x * (1 - cnt*w), where
// cnt = (#in-bounds 8-neighbors) = ry*rx - 1,
//   ry = 3 - (h==0) - (h==H-1),  rx = 3 - (w==0) - (w==W-1).
// W is a multiple of 4, so a float4 never crosses a row: only vector element 0
// can sit at w==0 and only element 3 at w==W-1.
// Bandwidth-bound: float4 non-temporal loads (stream-once, 256 MiB > L2),
// grid-stride, wave32 xor-reduce, one partial per block (scaled by `scale`).
// ---------------------------------------------------------------------------
__global__ __launch_bounds__(256) void smooth_partial_kernel(
    const float* __restrict__ x, const float* __restrict__ w,
    float* __restrict__ partials, int nvec4, float scale)
{
    __shared__ float wavesum[8];            // 256 threads = 8 waves (wave32)
    const int tid  = threadIdx.x;
    const int lane = tid & 31;
    const int wid  = tid >> 5;
    const int stride = gridDim.x * blockDim.x;

    float acc = 0.0f;
    for (int g = blockIdx.x * blockDim.x + tid; g < nvec4; g += stride) {
        const int e  = g << 2;              // flat element index over [C,H,W]
        const int i  = e & HW_MASK;         // per-channel flat index
        const int h  = i >> 12;             // /W  (W = 4096)
        const int w0 = i & (W_DIM - 1);     // column of element 0 (multiple of 4)

        const fv4 xv = __builtin_nontemporal_load((const fv4*)(x + e));
        const fv4 wv = __builtin_nontemporal_load((const fv4*)(w + e));

        const float ry  = 3.0f - (h == 0 ? 1.0f : 0.0f) - (h == H_DIM - 1 ? 1.0f : 0.0f);
        const float rx0 = (w0 == 0)         ? 2.0f : 3.0f;
        const float rx3 = (w0 == W_DIM - 4) ? 2.0f : 3.0f;
        const float c0  = fmaf(ry, rx0, -1.0f);   // cnt for element 0
        const float c12 = fmaf(ry, 3.0f, -1.0f);  // cnt for elements 1,2
        const float c3  = fmaf(ry, rx3, -1.0f);   // cnt for element 3

        acc = fmaf(xv.x, fmaf(-c0,  wv.x, 1.0f), acc);
        acc = fmaf(xv.y, fmaf(-c12, wv.y, 1.0f), acc);
        acc = fmaf(xv.z, fmaf(-c12, wv.z, 1.0f), acc);
        acc = fmaf(xv.w, fmaf(-c3,  wv.w, 1.0f), acc);
    }

    // wave32 butterfly reduction (deterministic)
    #pragma unroll
    for (int m = 16; m >= 1; m >>= 1)
        acc += __shfl_xor(acc, m, 32);

    if (lane == 0) wavesum[wid] = acc;
    __syncthreads();
    if (tid == 0) {
        float s = 0.0f;
        #pragma unroll
        for (int k = 0; k < 8; ++k) s += wavesum[k];
        partials[blockIdx.x] = s * scale;
    }
}

// ---------------------------------------------------------------------------
// Pass 2: one wave32 reduces the block partials through the f32 WMMA unit.
// Precondition: n64 is a multiple of 64 (guaranteed by kernel_launch), so the
// loads are unconditional 8-byte b64 loads — no exec masking around the WMMA.
// A (16x4 f32) holds 64 partials per iteration; packing is layout-agnostic
// because B = all-ones: D[m][n] = sum_k A[m][k] + C[m][n], and every A element
// lands in every column exactly once. Total = sum of column N=0 of C, which
// lives in lane 0 (M=0..7, VGPRs 0..7) and lane 16 (M=8..15, VGPRs 0..7).
// EXEC is all-1s for the whole wave (blockDim = 32), as WMMA requires.
// ---------------------------------------------------------------------------
__global__ __launch_bounds__(32) void smooth_final_kernel(
    const float* __restrict__ partials, int n64, float* __restrict__ out)
{
    const int lane = threadIdx.x;           // 0..31, all active
    v8f c = {};
    v2f b;
    b.x = 1.0f;
    b.y = 1.0f;                             // B = all-ones 4x16

    for (int base = 0; base < n64; base += 64) {
        const v2f a = *reinterpret_cast<const v2f*>(partials + base + 2 * lane);
        // (neg_a, A, neg_b, B, c_mod, C, reuse_a, reuse_b)
        c = __builtin_amdgcn_wmma_f32_16x16x4_f32(
            false, a, false, b, (short)0, c, false, false);
    }

    // per-lane sum over the 8 accumulator VGPRs = column-N entries M=0..7 (or 8..15)
    float s = c[0] + c[1] + c[2] + c[3] + c[4] + c[5] + c[6] + c[7];
    const float s_lo = __shfl(s, 0, 32);    // N=0, M=0..7
    const float s_hi = __shfl(s, 16, 32);   // N=0, M=8..15
    if (lane == 0) out[0] = (s_lo + s_hi) * INV_HW;   // exact scale: 1/2^24
}

extern "C" void kernel_launch(void* const* d_in, const int* in_sizes, int n_in,
                              void* d_out, int out_size, void* d_ws, size_t ws_size,
                              hipStream_t stream) {
    const float* x = (const float*)d_in[0];
    const float* w = (const float*)d_in[1];
    float* out = (float*)d_out;

    const size_t cap = ws_size / sizeof(float);

    if (cap < 64) {
        // Degenerate workspace: single block computes the whole (scaled) sum
        // directly into d_out. No WMMA pass needed for correctness.
        smooth_partial_kernel<<<1, 256, 0, stream>>>(x, w, out, NVEC4, INV_HW);
        return;
    }

    // blocks is ALWAYS a multiple of 64 so pass 2 runs guard-free.
    int blocks = 2048;                       // 2048 blocks x 8 waves
                                             // with b128 NT loads
    if ((size_t)blocks > cap) blocks = (int)(cap & ~(size_t)63);

    float* partials = (float*)d_ws;
    smooth_partial_kernel<<<blocks, 256, 0, stream>>>(x, w, partials, NVEC4, 1.0f);
    smooth_final_kernel<<<1, 32, 0, stream>>>(partials, blocks, out);
}